// coattention_21869973471940
// MI455X (gfx1250) — compile-verified
//
#include <hip/hip_runtime.h>

// CDNA5 (gfx1250) wave32 WMMA types
typedef __attribute__((ext_vector_type(2))) float v2f;  // A (16x4 f32) / B (4x16 f32): 2 VGPRs
typedef __attribute__((ext_vector_type(8))) float v8f;  // C/D (16x16 f32): 8 VGPRs

#define LC_  100
#define LS_  12
#define DD_  10

// One wave per batch element. The entire reference collapses (softmax over a
// size-1 axis == 1.0 exactly) to:
//   out[b, 0:10]  = sum_{l<12}  x1[b,l,:]
//   out[b,10:20]  = sum_{l<100} C [b,l,:]
// We compute each column-sum as ones(16x4) x data(4x16) accumulated with
// V_WMMA_F32_16X16X4_F32 on the matrix pipe; kernel stays memory-bound.
__global__ __launch_bounds__(256) void coatt_colsum_wmma_kernel(
    const float* __restrict__ Cin,
    const float* __restrict__ X1in,
    float* __restrict__ out,
    int Btot)
{
  const int wave = (blockIdx.x << 3) + (threadIdx.x >> 5);  // 8 wave32 per block
  if (wave >= Btot) return;  // wave-uniform: EXEC stays all-ones for live waves

  const int lane = threadIdx.x & 31;
  const int half = lane >> 4;   // 0: B rows {k, k+... }, 1: shifted row
  const int n    = lane & 15;   // B-matrix column (only n<10 is real data)

  // A-matrix = all ones (16x4 f32 -> 2 values per lane)
  v2f ones;
  ones.x = 1.0f;
  ones.y = 1.0f;

  // ---- cfinal: sum over the 100 rows of C[b] ------------------------------
  {
    const float* Cb = Cin + (size_t)wave * (LC_ * DD_);
    v8f acc = {};
#pragma unroll
    for (int l0 = 0; l0 < LC_; l0 += 4) {
      // B layout (4x16 f32): v0 holds rows {l0, l0+1}, v1 holds rows {l0+2, l0+3}
      int i0 = (l0 + half) * DD_ + n;
      int i1 = (l0 + 2 + half) * DD_ + n;
      // clamp so padding columns (n>=10) of the final chunk never leave the tensor
      i0 = i0 < LC_ * DD_ ? i0 : LC_ * DD_ - 1;
      i1 = i1 < LC_ * DD_ ? i1 : LC_ * DD_ - 1;
      v2f bm;
      bm.x = Cb[i0];
      bm.y = Cb[i1];
      acc = __builtin_amdgcn_wmma_f32_16x16x4_f32(
          /*neg_a=*/false, ones, /*neg_b=*/false, bm,
          /*c_mod=*/(short)0, acc, /*reuse_a=*/false, /*reuse_b=*/false);
    }
    // D row M=0 lives in lanes 0..15, accumulator VGPR 0: D[0,n] = sum_l C[l,n]
    if (lane < DD_) out[(size_t)wave * 20 + DD_ + lane] = acc[0];
  }

  // ---- sfinal: sum over the 12 rows of x1[b] ------------------------------
  {
    const float* Xb = X1in + (size_t)wave * (LS_ * DD_);
    v8f acc = {};
#pragma unroll
    for (int l0 = 0; l0 < LS_; l0 += 4) {
      int i0 = (l0 + half) * DD_ + n;
      int i1 = (l0 + 2 + half) * DD_ + n;
      i0 = i0 < LS_ * DD_ ? i0 : LS_ * DD_ - 1;
      i1 = i1 < LS_ * DD_ ? i1 : LS_ * DD_ - 1;
      v2f bm;
      bm.x = Xb[i0];
      bm.y = Xb[i1];
      acc = __builtin_amdgcn_wmma_f32_16x16x4_f32(
          false, ones, false, bm, (short)0, acc, false, false);
    }
    if (lane < DD_) out[(size_t)wave * 20 + lane] = acc[0];
  }
}

extern "C" void kernel_launch(void* const* d_in, const int* in_sizes, int n_in,
                              void* d_out, int out_size, void* d_ws, size_t ws_size,
                              hipStream_t stream) {
  (void)n_in; (void)d_ws; (void)ws_size; (void)out_size;
  const float* C  = (const float*)d_in[0];   // (B,100,10) f32
  const float* x1 = (const float*)d_in[1];   // (B,12,10)  f32
  float* out = (float*)d_out;                // (B,1,20)   f32

  const int Btot = in_sizes[0] / (LC_ * DD_);        // 16384
  const int waves_per_block = 256 / 32;              // 8
  const int blocks = (Btot + waves_per_block - 1) / waves_per_block;  // 2048

  coatt_colsum_wmma_kernel<<<blocks, 256, 0, stream>>>(C, x1, out, Btot);
}